// ChebNet_26010321944987
// MI455X (gfx1250) — compile-verified
//
#include <hip/hip_runtime.h>
#include <hip/hip_bf16.h>
#include <math.h>

// ---------------- problem constants (fixed by reference) ----------------
#define NN    10000
#define EE    640000
#define FIN   128
#define HID   128
#define CLS   16
#define KORD  4
#define MTILES 625            // 10000 / 16 exactly

typedef float v2f __attribute__((ext_vector_type(2)));
typedef float v8f __attribute__((ext_vector_type(8)));

// ---------------- tiny elementwise kernels ----------------
__global__ __launch_bounds__(256) void fill_zero(float* __restrict__ p, int n) {
    int i = blockIdx.x * 256 + threadIdx.x;
    if (i < n) p[i] = 0.0f;
}

__global__ __launch_bounds__(256) void neg_copy(float* __restrict__ dst,
                                                const float* __restrict__ src, int n) {
    int i = blockIdx.x * 256 + threadIdx.x;
    if (i < n) dst[i] = -src[i];
}

__global__ __launch_bounds__(256) void deg_kernel(const int* __restrict__ src,
                                                  float* __restrict__ deg) {
    int e = blockIdx.x * 256 + threadIdx.x;
    if (e < EE) atomicAdd(&deg[src[e]], 1.0f);
}

__global__ __launch_bounds__(256) void dis_kernel(const float* __restrict__ deg,
                                                  float* __restrict__ dis) {
    int i = blockIdx.x * 256 + threadIdx.x;
    if (i < NN) {
        float d = deg[i];
        dis[i] = (d > 0.0f) ? rsqrtf(fmaxf(d, 1e-12f)) : 0.0f;
    }
}

__global__ __launch_bounds__(256) void norm_kernel(const int* __restrict__ src,
                                                   const int* __restrict__ dst,
                                                   const float* __restrict__ dis,
                                                   float* __restrict__ norm) {
    int e = blockIdx.x * 256 + threadIdx.x;
    if (e < EE) norm[e] = -dis[src[e]] * dis[dst[e]];
}

// Transpose weights: Wt[k][n][f] = W[k][f][n] so the B-operand K-pair a lane
// needs is contiguous (one global_load_b64 instead of two b32).
__global__ __launch_bounds__(256) void transpose_w(const float* __restrict__ W,
                                                   float* __restrict__ Wt,
                                                   int F, int Ncols) {
    int i = blockIdx.x * 256 + threadIdx.x;          // over K*F*Ncols
    int tot = KORD * F * Ncols;
    if (i >= tot) return;
    int k = i / (F * Ncols);
    int r = i - k * (F * Ncols);
    int f = r / Ncols;
    int n = r - f * Ncols;
    Wt[(size_t)k * F * Ncols + (size_t)n * F + f] = W[i];
}

// ---------------- edge scatter (SpMM pass): out[dst] += scale*norm[e]*h[src] ----
// One wave32 per edge; lane covers features {lane, lane+32, lane+64, lane+96}.
// Gather of h row and atomic adds to out row are both fully coalesced.
__global__ __launch_bounds__(256) void scatter_prop(const float* __restrict__ h,
                                                    const int* __restrict__ src,
                                                    const int* __restrict__ dst,
                                                    const float* __restrict__ norm,
                                                    float scale,
                                                    float* __restrict__ out) {
    int wave = threadIdx.x >> 5;
    int lane = threadIdx.x & 31;
    int e = blockIdx.x * 8 + wave;
    if (e >= EE) return;
    int s = src[e];
    int d = dst[e];
    float w = scale * norm[e];
    const float* hs = h + (size_t)s * FIN;
    float* od = out + (size_t)d * FIN;
#pragma unroll
    for (int j = 0; j < 4; ++j) {
        int f = lane + 32 * j;
        atomicAdd(&od[f], w * hs[f]);
    }
}

// ---------------- layer-1 GEMM: H = relu(sum_k T_k @ W1[k] + b1) ----------------
// block = 256 threads = 8 waves; block handles a 16-row M-tile, wave w handles
// output columns [16w, 16w+16). A-tile staged in LDS (stride 132 -> bank clean).
// fp32 WMMA 16x16x4; B read as contiguous float2 from transposed weights.
__global__ __launch_bounds__(256) void gemm1_relu(const float* __restrict__ A0,
                                                  const float* __restrict__ A1,
                                                  const float* __restrict__ A2,
                                                  const float* __restrict__ A3,
                                                  const float* __restrict__ Wt,  // [4][128n][128f]
                                                  const float* __restrict__ bias,// [128]
                                                  float* __restrict__ out) {     // [N][128]
    __shared__ float As[16][132];
    const int mtile = blockIdx.x;              // 0..624
    const int wave  = threadIdx.x >> 5;        // 0..7 == ntile
    const int lane  = threadIdx.x & 31;
    const int mrow  = lane & 15;
    const int half  = lane >> 4;               // 0: K0/K1, 1: K2/K3
    const int ncol  = wave * 16 + mrow;

    v8f acc = {0.f, 0.f, 0.f, 0.f, 0.f, 0.f, 0.f, 0.f};
#pragma unroll
    for (int k = 0; k < KORD; ++k) {
        const float* Ak = (k == 0 ? A0 : k == 1 ? A1 : k == 2 ? A2 : A3)
                          + (size_t)mtile * 16 * FIN;
        __syncthreads();
        for (int i = threadIdx.x; i < 16 * FIN; i += 256)
            As[i >> 7][i & 127] = Ak[i];
        __syncthreads();
        const float* Wk = Wt + (size_t)k * (FIN * HID) + (size_t)ncol * FIN;
#pragma unroll
        for (int kk = 0; kk < FIN; kk += 4) {
            v2f a, b;
            a.x = As[mrow][kk + half * 2 + 0];
            a.y = As[mrow][kk + half * 2 + 1];
            b = *(const v2f*)(Wk + kk + half * 2);   // 8B-aligned contiguous pair
            acc = __builtin_amdgcn_wmma_f32_16x16x4_f32(
                false, a, false, b, (short)0, acc, false, false);
        }
    }
    float bv = bias[ncol];
#pragma unroll
    for (int r = 0; r < 8; ++r) {
        int m = mtile * 16 + r + half * 8;     // C/D layout: VGPR r -> M=r (+8 for hi half)
        float v = acc[r] + bv;
        out[(size_t)m * HID + ncol] = v > 0.0f ? v : 0.0f;
    }
}

// ---------------- layer-2 GEMM: logits = sum_k U_k @ W2[k] + b2 ----------------
// C=16 -> a single N-tile; one wave per 16-row M-tile, 8 M-tiles per block.
__global__ __launch_bounds__(256) void gemm2(const float* __restrict__ A0,
                                             const float* __restrict__ A1,
                                             const float* __restrict__ A2,
                                             const float* __restrict__ A3,
                                             const float* __restrict__ Wt,  // [4][16n][128f]
                                             const float* __restrict__ bias,// [16]
                                             float* __restrict__ out) {     // [N][16]
    const int wave  = threadIdx.x >> 5;
    const int lane  = threadIdx.x & 31;
    const int mtile = blockIdx.x * 8 + wave;
    if (mtile >= MTILES) return;               // wave-uniform; EXEC stays all-1s
    const int mrow = lane & 15;
    const int half = lane >> 4;
    const int ncol = mrow;

    v8f acc = {0.f, 0.f, 0.f, 0.f, 0.f, 0.f, 0.f, 0.f};
#pragma unroll
    for (int k = 0; k < KORD; ++k) {
        const float* Ak = (k == 0 ? A0 : k == 1 ? A1 : k == 2 ? A2 : A3)
                          + (size_t)mtile * 16 * HID;
        const float* Wk = Wt + (size_t)k * (HID * CLS) + (size_t)ncol * HID;
#pragma unroll
        for (int kk = 0; kk < HID; kk += 4) {
            v2f a, b;
            a = *(const v2f*)(Ak + mrow * HID + kk + half * 2);
            b = *(const v2f*)(Wk + kk + half * 2);
            acc = __builtin_amdgcn_wmma_f32_16x16x4_f32(
                false, a, false, b, (short)0, acc, false, false);
        }
    }
    float bv = bias[ncol];
#pragma unroll
    for (int r = 0; r < 8; ++r) {
        int m = mtile * 16 + r + half * 8;
        out[(size_t)m * CLS + ncol] = acc[r] + bv;
    }
}

// ---------------- row log_softmax over 16 classes ----------------
__global__ __launch_bounds__(256) void logsoftmax16(const float* __restrict__ in,
                                                    float* __restrict__ out) {
    int row = blockIdx.x * 256 + threadIdx.x;
    if (row >= NN) return;
    const float* p = in + (size_t)row * CLS;
    float v[CLS];
    float m = -INFINITY;
#pragma unroll
    for (int i = 0; i < CLS; ++i) { v[i] = p[i]; m = fmaxf(m, v[i]); }
    float s = 0.0f;
#pragma unroll
    for (int i = 0; i < CLS; ++i) s += __expf(v[i] - m);
    float lse = m + __logf(s);
    float* o = out + (size_t)row * CLS;
#pragma unroll
    for (int i = 0; i < CLS; ++i) o[i] = v[i] - lse;
}

// ---------------- host orchestration ----------------
extern "C" void kernel_launch(void* const* d_in, const int* in_sizes, int n_in,
                              void* d_out, int out_size, void* d_ws, size_t ws_size,
                              hipStream_t stream) {
    const float* x    = (const float*)d_in[0];                 // [N,128]
    const int*   ei   = (const int*)d_in[1];                   // [2,E]
    const float* W1   = (const float*)d_in[2];                 // [4,128,128]
    const float* b1   = (const float*)d_in[3];                 // [128]
    const float* W2   = (const float*)d_in[4];                 // [4,128,16]
    const float* b2   = (const float*)d_in[5];                 // [16]
    float*       outp = (float*)d_out;                         // [N,16]

    const int* srcI = ei;
    const int* dstI = ei + EE;

    // workspace layout (floats)
    float* ws   = (float*)d_ws;
    float* deg  = ws;                      // N
    float* dis  = deg  + NN;               // N
    float* nrm  = dis  + NN;               // E
    float* T1   = nrm  + EE;               // N*128
    float* T2   = T1   + (size_t)NN * FIN; // N*128
    float* T3   = T2   + (size_t)NN * FIN; // N*128
    float* H    = T3   + (size_t)NN * FIN; // N*128
    float* LG   = H    + (size_t)NN * FIN; // N*16
    float* Wt1  = LG   + (size_t)NN * CLS; // 4*128*128
    float* Wt2  = Wt1  + KORD * FIN * HID; // 4*16*128
    (void)in_sizes; (void)n_in; (void)out_size; (void)ws_size;

    const int NF       = NN * FIN;
    const int gNF      = (NF + 255) / 256;
    const int gN       = (NN + 255) / 256;
    const int gE       = (EE + 255) / 256;
    const int gScatter = (EE + 7) / 8;     // 8 waves/block, 1 edge/wave

    // --- weight transposes (once per launch; L2-resident afterwards) ---
    transpose_w<<<(KORD * FIN * HID + 255) / 256, 256, 0, stream>>>(W1, Wt1, FIN, HID);
    transpose_w<<<(KORD * HID * CLS + 255) / 256, 256, 0, stream>>>(W2, Wt2, HID, CLS);

    // --- graph normalization ---
    fill_zero<<<gN, 256, 0, stream>>>(deg, NN);
    deg_kernel<<<gE, 256, 0, stream>>>(srcI, deg);
    dis_kernel<<<gN, 256, 0, stream>>>(deg, dis);
    norm_kernel<<<gE, 256, 0, stream>>>(srcI, dstI, dis, nrm);

    // --- layer 1 Chebyshev basis: T1=L x; T2=2L T1 - x; T3=2L T2 - T1 ---
    fill_zero<<<gNF, 256, 0, stream>>>(T1, NF);
    scatter_prop<<<gScatter, 256, 0, stream>>>(x, srcI, dstI, nrm, 1.0f, T1);
    neg_copy<<<gNF, 256, 0, stream>>>(T2, x, NF);
    scatter_prop<<<gScatter, 256, 0, stream>>>(T1, srcI, dstI, nrm, 2.0f, T2);
    neg_copy<<<gNF, 256, 0, stream>>>(T3, T1, NF);
    scatter_prop<<<gScatter, 256, 0, stream>>>(T2, srcI, dstI, nrm, 2.0f, T3);

    // --- layer 1 fused GEMM + bias + ReLU ---
    gemm1_relu<<<MTILES, 256, 0, stream>>>(x, T1, T2, T3, Wt1, b1, H);

    // --- layer 2 Chebyshev basis over H (reuse T buffers) ---
    fill_zero<<<gNF, 256, 0, stream>>>(T1, NF);
    scatter_prop<<<gScatter, 256, 0, stream>>>(H, srcI, dstI, nrm, 1.0f, T1);
    neg_copy<<<gNF, 256, 0, stream>>>(T2, H, NF);
    scatter_prop<<<gScatter, 256, 0, stream>>>(T1, srcI, dstI, nrm, 2.0f, T2);
    neg_copy<<<gNF, 256, 0, stream>>>(T3, T1, NF);
    scatter_prop<<<gScatter, 256, 0, stream>>>(T2, srcI, dstI, nrm, 2.0f, T3);

    // --- layer 2 GEMM + bias ---
    gemm2<<<(MTILES + 7) / 8, 256, 0, stream>>>(H, T1, T2, T3, Wt2, b2, LG);

    // --- log_softmax ---
    logsoftmax16<<<gN, 256, 0, stream>>>(LG, outp);
}